// DynamicSimAMWith4x4Slicing_87170656239668
// MI455X (gfx1250) — compile-verified
//
#include <hip/hip_runtime.h>
#include <math.h>

typedef __attribute__((ext_vector_type(2))) float v2f;
typedef __attribute__((ext_vector_type(4))) float v4f;
typedef __attribute__((ext_vector_type(8))) float v8f;

#define BB 8
#define CC 256
#define HH 256
#define WW 256
#define NBLK 32768   /* B*4*4*C */
#define NSITE 128    /* B*4*4  */

/* ws layout in floats */
#define WS_BLOCKSUM 0
#define WS_BLOCKD2  32768
#define WS_S        65536
#define WS_G        98304
#define WS_MISC     131072

__device__ __forceinline__ float sigmoidf_(float y) {
    return 1.0f / (1.0f + expf(-y));
}

/* blk = ((b*4+i)*4+j)*C + c ; returns element offset of block origin in x/out */
__device__ __forceinline__ size_t block_origin(int blk) {
    int c = blk & 255;
    int site = blk >> 8;
    int j = site & 3;
    int i = (site >> 2) & 3;
    int b = site >> 4;
    return (((size_t)(b * CC + c)) * HH + (size_t)i * 64) * WW + (size_t)j * 64;
}

__device__ __forceinline__ float block_tree_reduce(float v, float* sm) {
    int t = threadIdx.x;
    sm[t] = v;
    __syncthreads();
    #pragma unroll
    for (int k = 128; k > 0; k >>= 1) {
        if (t < k) sm[t] += sm[t + k];
        __syncthreads();
    }
    float r = sm[0];
    __syncthreads();
    return r;
}

/* ---- K1: per-block sum and sum of squared deviations ---- */
__global__ void __launch_bounds__(256) stats_kernel(const float* __restrict__ x,
                                                    float* __restrict__ ws) {
    const int blk = blockIdx.x;
    const int t = threadIdx.x;
    const float* base = x + block_origin(blk);
    float sum = 0.f, sq = 0.f;
    #pragma unroll
    for (int it = 0; it < 4; ++it) {
        int idx = it * 256 + t;          /* 1024 float4 slots */
        int row = idx >> 4, q = idx & 15;
        v4f v = *reinterpret_cast<const v4f*>(base + (size_t)row * WW + q * 4);
        sum += v.x + v.y + v.z + v.w;
        sq  += v.x * v.x + v.y * v.y + v.z * v.z + v.w * v.w;
    }
    __shared__ float sm[256];
    float tsum = block_tree_reduce(sum, sm);
    float tsq  = block_tree_reduce(sq, sm);
    if (t == 0) {
        ws[WS_BLOCKSUM + blk] = tsum;
        ws[WS_BLOCKD2 + blk]  = tsq - tsum * tsum * (1.0f / 4096.0f); /* sum of (x-mu)^2 */
    }
}

/* ---- K2: reduce block sums -> dyn_lambda ---- */
__global__ void __launch_bounds__(1024) lambda_kernel(float* __restrict__ ws) {
    __shared__ double sm[1024];
    const int t = threadIdx.x;
    double s = 0.0;
    for (int idx = t; idx < NBLK; idx += 1024) s += (double)ws[WS_BLOCKSUM + idx];
    sm[t] = s;
    __syncthreads();
    #pragma unroll
    for (int k = 512; k > 0; k >>= 1) {
        if (t < k) sm[t] += sm[t + k];
        __syncthreads();
    }
    if (t == 0) {
        double mean = sm[0] / ((double)BB * CC * HH * WW);
        ws[WS_MISC] = (float)(1e-4 * log1p(fabs(mean)));
    }
}

/* ---- K3: per-block s_c = mean(x * sigmoid(y)) ---- */
__global__ void __launch_bounds__(256) s_kernel(const float* __restrict__ x,
                                                float* __restrict__ ws) {
    const int blk = blockIdx.x;
    const int t = threadIdx.x;
    const float* base = x + block_origin(blk);
    const float lam  = ws[WS_MISC];
    const float mu   = ws[WS_BLOCKSUM + blk] * (1.0f / 4096.0f);
    const float coef = 1.0f / (4.0f * (ws[WS_BLOCKD2 + blk] * (1.0f / 4095.0f) + lam));
    float acc = 0.f;
    #pragma unroll
    for (int it = 0; it < 4; ++it) {
        int idx = it * 256 + t;
        int row = idx >> 4, q = idx & 15;
        v4f v = *reinterpret_cast<const v4f*>(base + (size_t)row * WW + q * 4);
        float d;
        d = v.x - mu; acc += v.x * sigmoidf_(d * d * coef + 0.5f);
        d = v.y - mu; acc += v.y * sigmoidf_(d * d * coef + 0.5f);
        d = v.z - mu; acc += v.z * sigmoidf_(d * d * coef + 0.5f);
        d = v.w - mu; acc += v.w * sigmoidf_(d * d * coef + 0.5f);
    }
    __shared__ float sm[256];
    float tot = block_tree_reduce(acc, sm);
    if (t == 0) ws[WS_S + blk] = tot * (1.0f / 4096.0f);
}

/* ---- K4: SE attention via fp32 WMMA.
   G[128x256] = sigmoid( relu( S[128x256] @ w1^T[256x16] ) @ w2^T[16x256] )
   One workgroup, 8 waves; wave w owns M-tile rows [16w, 16w+16). ---- */
__global__ void __launch_bounds__(256) se_wmma_kernel(const float* __restrict__ w1,
                                                      const float* __restrict__ w2,
                                                      float* __restrict__ ws) {
    __shared__ float Hs[NSITE][17];  /* H[128x16], padded */
    const int t = threadIdx.x;
    const int wave = t >> 5;
    const int lane = t & 31;
    const int n = lane & 15;          /* A: row M offset; B: column N; C/D: column N */
    const int khalf = lane >> 4;      /* lane half selects K pair / M+8 rows */
    const int m0 = wave * 16;
    const float* S = ws + WS_S;
    float* G = ws + WS_G;

    /* GEMM1: H = relu(S @ w1^T), K = 256 -> 64 wmma steps of K=4 */
    v8f acc = {};
    const float* arow = S + (size_t)(m0 + n) * 256;
    const float* brow = w1 + (size_t)n * 256;   /* w1[n][k] = (w1^T)[k][n] */
    #pragma unroll 8
    for (int k4 = 0; k4 < 64; ++k4) {
        const int kb = k4 * 4 + khalf * 2;
        v2f a, b;
        a.x = arow[kb]; a.y = arow[kb + 1];
        b.x = brow[kb]; b.y = brow[kb + 1];
        acc = __builtin_amdgcn_wmma_f32_16x16x4_f32(
            false, a, false, b, (short)0, acc, false, false);
    }
    #pragma unroll
    for (int r = 0; r < 8; ++r) {
        float h = acc[r];
        Hs[m0 + r + khalf * 8][n] = h > 0.0f ? h : 0.0f;  /* relu */
    }
    __syncthreads();

    /* GEMM2: G = sigmoid(H @ w2^T), K = 16 -> 4 wmma steps; 16 N-tiles */
    for (int nt = 0; nt < 16; ++nt) {
        v8f acc2 = {};
        #pragma unroll
        for (int k4 = 0; k4 < 4; ++k4) {
            const int kb = k4 * 4 + khalf * 2;
            v2f a, b;
            a.x = Hs[m0 + n][kb];  a.y = Hs[m0 + n][kb + 1];
            b.x = w2[(size_t)(nt * 16 + n) * 16 + kb];       /* w2[c][r] = (w2^T)[r][c] */
            b.y = w2[(size_t)(nt * 16 + n) * 16 + kb + 1];
            acc2 = __builtin_amdgcn_wmma_f32_16x16x4_f32(
                false, a, false, b, (short)0, acc2, false, false);
        }
        #pragma unroll
        for (int r = 0; r < 8; ++r) {
            const int M = m0 + r + khalf * 8;
            G[(size_t)M * 256 + nt * 16 + n] = sigmoidf_(acc2[r]);
        }
    }
}

/* ---- K5: out = eb * g, streaming stores ---- */
__global__ void __launch_bounds__(256) out_kernel(const float* __restrict__ x,
                                                  const float* __restrict__ ws,
                                                  float* __restrict__ out) {
    const int blk = blockIdx.x;
    const int t = threadIdx.x;
    const size_t org = block_origin(blk);
    const float* base = x + org;
    float* obase = out + org;
    const float lam  = ws[WS_MISC];
    const float mu   = ws[WS_BLOCKSUM + blk] * (1.0f / 4096.0f);
    const float coef = 1.0f / (4.0f * (ws[WS_BLOCKD2 + blk] * (1.0f / 4095.0f) + lam));
    const float g = ws[WS_G + blk];
    #pragma unroll
    for (int it = 0; it < 4; ++it) {
        int idx = it * 256 + t;
        int row = idx >> 4, q = idx & 15;
        const size_t off = (size_t)row * WW + q * 4;
        v4f v = *reinterpret_cast<const v4f*>(base + off);
        v4f o; float d;
        d = v.x - mu; o.x = v.x * sigmoidf_(d * d * coef + 0.5f) * g;
        d = v.y - mu; o.y = v.y * sigmoidf_(d * d * coef + 0.5f) * g;
        d = v.z - mu; o.z = v.z * sigmoidf_(d * d * coef + 0.5f) * g;
        d = v.w - mu; o.w = v.w * sigmoidf_(d * d * coef + 0.5f) * g;
        __builtin_nontemporal_store(o, reinterpret_cast<v4f*>(obase + off));
    }
}

extern "C" void kernel_launch(void* const* d_in, const int* in_sizes, int n_in,
                              void* d_out, int out_size, void* d_ws, size_t ws_size,
                              hipStream_t stream) {
    (void)in_sizes; (void)n_in; (void)out_size; (void)ws_size;
    const float* x  = (const float*)d_in[0];
    const float* w1 = (const float*)d_in[1];
    const float* w2 = (const float*)d_in[2];
    float* out = (float*)d_out;
    float* ws  = (float*)d_ws;

    stats_kernel  <<<NBLK, 256, 0, stream>>>(x, ws);
    lambda_kernel <<<1, 1024, 0, stream>>>(ws);
    s_kernel      <<<NBLK, 256, 0, stream>>>(x, ws);
    se_wmma_kernel<<<1, 256, 0, stream>>>(w1, w2, ws);
    out_kernel    <<<NBLK, 256, 0, stream>>>(x, ws, out);
}